// NodeModel_TWO_10393820857012
// MI455X (gfx1250) — compile-verified
//
#include <hip/hip_runtime.h>

// NodeModel GNN layer for MI455X (gfx1250, wave32, WMMA).
//
//   msg = concat(x[row], edge_attr) @ W1 + b1        [E=640000, 160] -> [E,128]
//   agg = segment_sum(msg, col, N)                   atomic f32 scatter (L2-resident)
//   out = concat(x, agg) @ W2 + b2                   [N=200000, 256] -> [N,128]
//
// fp32 WMMA (V_WMMA_F32_16X16X4_F32) keeps exact reference precision; the
// problem is compute/atomic bound (39 GFLOP vs ~13us of HBM traffic), so the
// matrix pipe is the right execution unit even at K=4 per instruction.

#define N_NODES 200000
#define N_EDGES 640000
#define D_NODE  128
#define D_EDGE  32
#define HIDDEN  128
#define D_IN1   160   // D_NODE + D_EDGE
#define D_IN2   256   // D_NODE + HIDDEN

// LDS row strides (in floats) padded so 16 lanes reading a column hit 16
// distinct banks: 164 % 64 = 36 (gcd(36,64)=4, 16 rows -> distinct banks),
// 260 % 64 = 4 (16 rows -> banks 0,4,...,60).
#define AS1 164
#define AS2 260

typedef __attribute__((ext_vector_type(2))) float v2f;
typedef __attribute__((ext_vector_type(4))) float v4f;
typedef __attribute__((ext_vector_type(8))) float v8f;

// ---------------------------------------------------------------------------
// Zero d_out (it doubles as the agg accumulator for the atomic scatter).
// ---------------------------------------------------------------------------
__global__ void zero_out_kernel(float* __restrict__ out, int n4) {
  int i = blockIdx.x * blockDim.x + threadIdx.x;
  if (i < n4) {
    v4f z = {0.f, 0.f, 0.f, 0.f};
    ((v4f*)out)[i] = z;
  }
}

// ---------------------------------------------------------------------------
// Stage 1+2: per-edge GEMM tile + atomic scatter into agg (= d_out).
// One wave owns a 16-edge tile; A = [16 x 160] staged in LDS (coalesced),
// B fragments streamed from W1 (80 KB, L2/WGP$ resident).
// ---------------------------------------------------------------------------
__global__ __launch_bounds__(64) void edge_mm_scatter_kernel(
    const float* __restrict__ x, const int* __restrict__ eidx,
    const float* __restrict__ ea, const float* __restrict__ W1,
    const float* __restrict__ b1, float* __restrict__ agg) {
  __shared__ float aS[2][16 * AS1];

  const int lane = threadIdx.x & 31;
  const int wave = threadIdx.x >> 5;
  const int e0   = (blockIdx.x * 2 + wave) * 16;
  float* a = aS[wave];

  // Stage A tile: row r = [ x[src(e0+r)][0..127] | edge_attr[e0+r][0..31] ].
  // 32 consecutive lanes load 32 consecutive floats -> 128B coalesced.
  for (int r = 0; r < 16; ++r) {
    const int src = eidx[e0 + r];  // edge_index[0][e] (source node)
    const float* xr = x  + (long long)src * D_NODE;
    const float* er = ea + (long long)(e0 + r) * D_EDGE;
    for (int c = lane; c < D_IN1; c += 32)
      a[r * AS1 + c] = (c < D_NODE) ? xr[c] : er[c - D_NODE];
  }
  // Wave-private LDS slice: DS ops are in-order within the wave, no barrier.

  const int m16  = lane & 15;
  const int kb   = (lane >> 4) << 1;  // lane half selects K {0,1} vs {2,3}
  const int mOff = (lane >> 4) << 3;  // C/D: lanes 16-31 hold rows M = r+8

  for (int nt = 0; nt < 8; ++nt) {
    const int col = nt * 16 + m16;
    const float bb = b1[col];
    v8f acc = {bb, bb, bb, bb, bb, bb, bb, bb};  // bias broadcast per column

    for (int k = 0; k < D_IN1; k += 4) {
      v2f af, bf;
      af.x = a[m16 * AS1 + k + kb];                 // A: M=m16, K=k+kb
      af.y = a[m16 * AS1 + k + kb + 1];             //    K=k+kb+1
      bf.x = W1[(k + kb)     * HIDDEN + col];       // B: K=k+kb,  N=col
      bf.y = W1[(k + kb + 1) * HIDDEN + col];       //    K=k+kb+1 (64B/half-wave)
      acc = __builtin_amdgcn_wmma_f32_16x16x4_f32(
          /*neg_a=*/false, af, /*neg_b=*/false, bf,
          /*c_mod=*/(short)0, acc, /*reuse_a=*/false, /*reuse_b=*/false);
    }

    // Scatter-add this 16x16 tile by destination node (edge_index[1]).
    // Atomics land in L2 (agg region = 102 MB < 192 MB L2).
    for (int r = 0; r < 8; ++r) {
      const int M   = r + mOff;
      const int dst = eidx[N_EDGES + e0 + M];
      unsafeAtomicAdd(&agg[(long long)dst * HIDDEN + col], acc[r]);
    }
  }
}

// ---------------------------------------------------------------------------
// Stage 3: out = concat(x, agg) @ W2 + b2, computed IN PLACE on d_out.
// Each wave stages its 16 rows (x | agg) into LDS before overwriting them.
// ---------------------------------------------------------------------------
__global__ __launch_bounds__(64) void node_mm_kernel(
    const float* __restrict__ x, const float* __restrict__ W2,
    const float* __restrict__ b2, float* __restrict__ out) {
  __shared__ float aS[2][16 * AS2];

  const int lane = threadIdx.x & 31;
  const int wave = threadIdx.x >> 5;
  const int n0   = (blockIdx.x * 2 + wave) * 16;
  float* a = aS[wave];

  // Stage A tile: row r = [ x[n0+r][0..127] | agg[n0+r][0..127] ].
  for (int r = 0; r < 16; ++r) {
    const float* xr = x   + (long long)(n0 + r) * D_NODE;
    const float* gr = out + (long long)(n0 + r) * HIDDEN;   // agg lives in d_out
    for (int c = lane; c < D_IN2; c += 32)
      a[r * AS2 + c] = (c < D_NODE) ? xr[c] : gr[c - D_NODE];
  }

  const int m16  = lane & 15;
  const int kb   = (lane >> 4) << 1;
  const int mOff = (lane >> 4) << 3;

  for (int nt = 0; nt < 8; ++nt) {
    const int col = nt * 16 + m16;
    const float bb = b2[col];
    v8f acc = {bb, bb, bb, bb, bb, bb, bb, bb};

    for (int k = 0; k < D_IN2; k += 4) {
      v2f af, bf;
      af.x = a[m16 * AS2 + k + kb];
      af.y = a[m16 * AS2 + k + kb + 1];
      bf.x = W2[(k + kb)     * D_NODE + col];
      bf.y = W2[(k + kb + 1) * D_NODE + col];
      acc = __builtin_amdgcn_wmma_f32_16x16x4_f32(
          false, af, false, bf, (short)0, acc, false, false);
    }

    for (int r = 0; r < 8; ++r)
      out[(long long)(n0 + r + mOff) * D_NODE + col] = acc[r];
  }
}

// ---------------------------------------------------------------------------
// Launch: zero agg -> edge GEMM + scatter -> node GEMM (in place on d_out).
// Inputs (setup_inputs order): x, edge_index, edge_attr, u(unused),
// batch(unused), W1, b1, W2, b2.  edge_index is JAX-canonicalized int32.
// ---------------------------------------------------------------------------
extern "C" void kernel_launch(void* const* d_in, const int* in_sizes, int n_in,
                              void* d_out, int out_size, void* d_ws, size_t ws_size,
                              hipStream_t stream) {
  (void)in_sizes; (void)n_in; (void)out_size; (void)d_ws; (void)ws_size;
  const float* x    = (const float*)d_in[0];
  const int*   eidx = (const int*)  d_in[1];
  const float* ea   = (const float*)d_in[2];
  const float* W1   = (const float*)d_in[5];
  const float* b1   = (const float*)d_in[6];
  const float* W2   = (const float*)d_in[7];
  const float* b2   = (const float*)d_in[8];
  float* out = (float*)d_out;

  const int n4 = N_NODES * D_NODE / 4;  // 6.4M float4
  zero_out_kernel<<<(n4 + 255) / 256, 256, 0, stream>>>(out, n4);

  // 640000 edges / (16 per wave * 2 waves per block) = 20000 blocks
  edge_mm_scatter_kernel<<<N_EDGES / 32, 64, 0, stream>>>(x, eidx, ea, W1, b1, out);

  // 200000 nodes / 32 = 6250 blocks
  node_mm_kernel<<<N_NODES / 32, 64, 0, stream>>>(x, W2, b2, out);
}